// BayesianSG_81020263072021
// MI455X (gfx1250) — compile-verified
//
#include <hip/hip_runtime.h>
#include <hip/hip_bf16.h>
#include <math.h>

// ---------------------------------------------------------------------------
// Bayesian skip-gram forward for MI455X (gfx1250, wave32).
// Dominant work: z[1024,128] @ W_vocab[128,50257] (13.2 GFLOP, fp32) with
// streaming online-softmax (never materializes the 205MB logits matrix).
// Matrix math uses V_WMMA_F32_16X16X4_F32 (native fp32 WMMA path).
// Hot loop: A-fragments register-resident, per-lane online softmax (no
// shuffles in the loop; one log-sum-exp lane merge at the end).
// ---------------------------------------------------------------------------

#define DIM      128
#define TWO_D    256
#define BQ       1024
#define CCTX     10
#define VOCAB    50257
#define NTILES   ((VOCAB + 15) / 16)   // 3142 vocab tiles of 16
#define NSLICE   32                    // vocab slices (== wave32 lanes for combine)
#define TPS      ((NTILES + NSLICE - 1) / NSLICE)  // 99 tiles per slice

typedef __attribute__((ext_vector_type(2))) float v2f;
typedef __attribute__((ext_vector_type(8))) float v8f;

#define NEG_INF  (-__builtin_inff())
#define BIG_NEG  (-3.0e38f)   // finite "running max" init: avoids inf-inf NaN

__device__ inline v8f wmma_f32_16x16x4(v2f a, v2f b, v8f c) {
  // 8 args: (neg_a, A, neg_b, B, c_mod, C, reuse_a, reuse_b)
  return __builtin_amdgcn_wmma_f32_16x16x4_f32(false, a, false, b, (short)0, c,
                                               false, false);
}

__device__ inline float softplusf(float x) {
  // numerically stable: max(x,0) + log1p(exp(-|x|))
  return fmaxf(x, 0.f) + log1pf(__expf(-fabsf(x)));
}

// ---------------------------------------------------------------------------
// Kernel 1: h[b,:] = sum_c relu(concat(ce, cx_c) @ W_enc + b_enc)
// One block per batch row. 16-row A tile (10 real context rows + 6 zero pad),
// 16 waves each produce a 16x16 output tile over K=256 (64 WMMAs).
// ---------------------------------------------------------------------------
__global__ __launch_bounds__(512) void enc_kernel(
    const int* __restrict__ center_id, const int* __restrict__ ctx_ids,
    const float* __restrict__ emb, const float* __restrict__ W_enc,
    const float* __restrict__ b_enc, float* __restrict__ h_out) {
  __shared__ float lds_a[16 * 260];  // stride 260: 4-bank row advance, b64 clean
  const int b = blockIdx.x;
  const int tid = threadIdx.x;
  const int cid = center_id[b];

  for (int idx = tid; idx < 16 * TWO_D; idx += 512) {
    const int row = idx >> 8;
    const int col = idx & 255;
    float v = 0.f;
    if (row < CCTX) {
      if (col < DIM) {
        v = emb[(size_t)cid * DIM + col];
      } else {
        const int cx = ctx_ids[b * CCTX + row];
        v = emb[(size_t)cx * DIM + (col - DIM)];
      }
    }
    lds_a[row * 260 + col] = v;
  }
  __syncthreads();

  const int wave = tid >> 5;         // 0..15 -> N chunk
  const int lane = tid & 31;
  const int m = lane & 15;           // A row
  const int kh = (lane < 16) ? 0 : 2;
  const int n = wave * 16 + (lane & 15);
  const float bias = b_enc[n];

  v8f acc = {};
#pragma unroll 4
  for (int kb = 0; kb < 64; ++kb) {
    const int k0 = kb * 4 + kh;
    v2f a, bb;
    a.x = lds_a[m * 260 + k0];
    a.y = lds_a[m * 260 + k0 + 1];
    bb.x = W_enc[(size_t)k0 * TWO_D + n];
    bb.y = W_enc[(size_t)(k0 + 1) * TWO_D + n];
    acc = wmma_f32_16x16x4(a, bb, acc);
  }

  // D layout: VGPR r holds row r (lanes 0-15) / row r+8 (lanes 16-31), col n.
  // Sum relu over valid rows 0..9 only (pad rows excluded).
  float s;
  if (lane < 16) {
    s = 0.f;
#pragma unroll
    for (int r = 0; r < 8; ++r) s += fmaxf(acc[r] + bias, 0.f);  // rows 0..7
  } else {
    s = fmaxf(acc[0] + bias, 0.f) + fmaxf(acc[1] + bias, 0.f);   // rows 8,9
  }
  s += __shfl_xor(s, 16, 32);
  if (lane < 16) h_out[(size_t)b * TWO_D + n] = s;
}

// ---------------------------------------------------------------------------
// Kernel 2: mean/var heads, reparameterized z, diagonal-Gaussian KL per row.
// Tiny (0.13 GFLOP) -> plain VALU. One block of 128 threads per batch row.
// ---------------------------------------------------------------------------
__global__ __launch_bounds__(128) void latent_kernel(
    const int* __restrict__ center_id, const float* __restrict__ eps,
    const float* __restrict__ prior_means, const float* __restrict__ prior_vars,
    const float* __restrict__ W_mean, const float* __restrict__ b_mean,
    const float* __restrict__ W_var, const float* __restrict__ b_var,
    const float* __restrict__ h_in, float* __restrict__ z_out,
    float* __restrict__ kl_out) {
  __shared__ float hs[TWO_D];
  __shared__ float red[4];
  const int b = blockIdx.x, t = threadIdx.x;
  hs[t] = h_in[(size_t)b * TWO_D + t];
  hs[t + 128] = h_in[(size_t)b * TWO_D + t + 128];
  __syncthreads();

  float mean = b_mean[t], vl = b_var[t];
  for (int k = 0; k < TWO_D; ++k) {
    const float hk = hs[k];
    mean = fmaf(hk, W_mean[(size_t)k * DIM + t], mean);
    vl = fmaf(hk, W_var[(size_t)k * DIM + t], vl);
  }
  const float var = softplusf(vl);
  z_out[(size_t)b * DIM + t] = mean + __expf(var * 0.5f) * eps[t];

  const int cid = center_id[b];
  const float pm = prior_means[(size_t)cid * DIM + t];
  const float pv = softplusf(prior_vars[(size_t)cid * DIM + t]);
  const float diff = pm - mean;
  float term = var / pv + diff * diff / pv - 1.f + __logf(pv) - __logf(var);
  for (int off = 16; off; off >>= 1) term += __shfl_xor(term, off, 32);
  if ((t & 31) == 0) red[t >> 5] = term;
  __syncthreads();
  if (t == 0) kl_out[b] = 0.5f * (red[0] + red[1] + red[2] + red[3]);
}

// ---------------------------------------------------------------------------
// Kernel 3: streaming logits + online-softmax partials.
// grid = (B/16 row-tiles, NSLICE vocab slices). 8 waves/block, each wave walks
// 16-wide vocab tiles. Per tile: 32 f32 WMMAs (A-fragments register-resident,
// loaded from LDS once), then a shuffle-free per-lane online (max,sumexp)
// update. Cross-lane log-sum-exp merge happens once after the loop.
// W_vocab stays L2-resident across row-tiles (25.7MB << 192MB L2).
// ---------------------------------------------------------------------------
__global__ __launch_bounds__(256) void vocab_lse_kernel(
    const float* __restrict__ z, const float* __restrict__ W_vocab,
    const float* __restrict__ b_vocab, float* __restrict__ part_m,
    float* __restrict__ part_s) {
  __shared__ float zt[16 * 132];
  __shared__ float cmb_m[8][16];
  __shared__ float cmb_s[8][16];
  const int b0 = blockIdx.x * 16;
  const int slice = blockIdx.y;
  const int tid = threadIdx.x;

  for (int idx = tid; idx < 16 * DIM; idx += 256) {
    const int row = idx >> 7, col = idx & 127;
    zt[row * 132 + col] = z[(size_t)(b0 + row) * DIM + col];
  }
  __syncthreads();

  const int wave = tid >> 5, lane = tid & 31;
  const int m = lane & 15;
  const int kh = (lane < 16) ? 0 : 2;

  // Hoist all 32 A-fragments (K=128 in steps of 4) into registers: the z tile
  // is invariant across the whole vocab sweep. 64 VGPRs.
  v2f a_frag[32];
#pragma unroll
  for (int kb = 0; kb < 32; ++kb) {
    const int k0 = kb * 4 + kh;
    a_frag[kb].x = zt[m * 132 + k0];
    a_frag[kb].y = zt[m * 132 + k0 + 1];
  }

  // Per-lane running (max, sumexp): each lane owns a disjoint column subset
  // across tiles, so no shuffles needed inside the loop.
  float m_run[8], s_run[8];
#pragma unroll
  for (int r = 0; r < 8; ++r) { m_run[r] = BIG_NEG; s_run[r] = 0.f; }

  const int t0 = slice * TPS;
  const int t1 = (t0 + TPS < NTILES) ? (t0 + TPS) : NTILES;
  for (int t = t0 + wave; t < t1; t += 8) {
    const int v0 = t * 16;
    const int v = v0 + (lane & 15);
    const int vc = (v < VOCAB) ? v : (VOCAB - 1);  // clamp tail loads
    const float* __restrict__ wp = W_vocab + vc;
    v8f acc = {};
#pragma unroll
    for (int kb = 0; kb < 32; ++kb) {
      const int k0 = kb * 4 + kh;
      v2f bb;
      bb.x = wp[(size_t)k0 * VOCAB];
      bb.y = wp[(size_t)(k0 + 1) * VOCAB];
      acc = wmma_f32_16x16x4(a_frag[kb], bb, acc);
    }
    const float bias = (v < VOCAB) ? b_vocab[v] : 0.f;
#pragma unroll
    for (int r = 0; r < 8; ++r) {
      // VGPR r = tile row r (lanes 0-15) / row r+8 (lanes 16-31); col = lane&15
      // Pad columns get -inf logits: exp(-inf - finite) == 0, max unaffected
      // (m_run init is finite BIG_NEG so rescale is never NaN).
      const float logit = (v < VOCAB) ? (acc[r] + bias) : NEG_INF;
      const float m_new = fmaxf(m_run[r], logit);
      s_run[r] = s_run[r] * __expf(m_run[r] - m_new) + __expf(logit - m_new);
      m_run[r] = m_new;
    }
  }

  // Cross-lane merge of (m,s) over the 16 columns of each half-wave.
#pragma unroll
  for (int r = 0; r < 8; ++r) {
    float mm = m_run[r], ss = s_run[r];
#pragma unroll
    for (int off = 1; off <= 8; off <<= 1) {
      const float mo = __shfl_xor(mm, off, 32);
      const float so = __shfl_xor(ss, off, 32);
      const float M = fmaxf(mm, mo);
      ss = ss * __expf(mm - M) + so * __expf(mo - M);
      mm = M;
    }
    m_run[r] = mm;
    s_run[r] = ss;
  }

  if ((lane & 15) == 0) {
    const int half = lane >> 4;
#pragma unroll
    for (int r = 0; r < 8; ++r) {
      cmb_m[wave][half * 8 + r] = m_run[r];
      cmb_s[wave][half * 8 + r] = s_run[r];
    }
  }
  __syncthreads();
  if (tid < 16) {
    float M = BIG_NEG;
    for (int w = 0; w < 8; ++w) M = fmaxf(M, cmb_m[w][tid]);
    float S = 0.f;
    for (int w = 0; w < 8; ++w) S += cmb_s[w][tid] * __expf(cmb_m[w][tid] - M);
    part_m[(size_t)slice * BQ + b0 + tid] = M;
    part_s[(size_t)slice * BQ + b0 + tid] = S;
  }
}

// ---------------------------------------------------------------------------
// Kernel 4: combine slice partials -> lse[b]; recompute the 10 context logits;
// resid[b] = rec[b] - kl[b]. One wave per batch row (NSLICE == 32 lanes).
// ---------------------------------------------------------------------------
__global__ __launch_bounds__(32) void rec_kernel(
    const int* __restrict__ ctx_ids, const float* __restrict__ z,
    const float* __restrict__ W_vocab, const float* __restrict__ b_vocab,
    const float* __restrict__ part_m, const float* __restrict__ part_s,
    const float* __restrict__ kl, float* __restrict__ resid) {
  const int b = blockIdx.x, lane = threadIdx.x;
  const float pm = part_m[(size_t)lane * BQ + b];
  const float ps = part_s[(size_t)lane * BQ + b];
  float M = pm;
  for (int off = 16; off; off >>= 1) M = fmaxf(M, __shfl_xor(M, off, 32));
  float S = ps * __expf(pm - M);
  for (int off = 16; off; off >>= 1) S += __shfl_xor(S, off, 32);
  const float lse = M + __logf(S);

  int ctx[CCTX];
#pragma unroll
  for (int k = 0; k < CCTX; ++k) ctx[k] = ctx_ids[b * CCTX + k];

  float acc = 0.f;
  for (int d = lane; d < DIM; d += 32) {
    const float zd = z[(size_t)b * DIM + d];
#pragma unroll
    for (int k = 0; k < CCTX; ++k)
      acc = fmaf(zd, W_vocab[(size_t)d * VOCAB + ctx[k]], acc);
  }
  if (lane < CCTX) acc += b_vocab[ctx_ids[b * CCTX + lane]];
  for (int off = 16; off; off >>= 1) acc += __shfl_xor(acc, off, 32);
  if (lane == 0) resid[b] = (acc - (float)CCTX * lse) - kl[b];
}

// ---------------------------------------------------------------------------
// Kernel 5: deterministic tree mean over B rows -> scalar output.
// ---------------------------------------------------------------------------
__global__ __launch_bounds__(256) void mean_kernel(const float* __restrict__ resid,
                                                   float* __restrict__ out) {
  __shared__ float red[256];
  const int t = threadIdx.x;
  red[t] = resid[t] + resid[t + 256] + resid[t + 512] + resid[t + 768];
  __syncthreads();
  for (int off = 128; off; off >>= 1) {
    if (t < off) red[t] += red[t + off];
    __syncthreads();
  }
  if (t == 0) out[0] = red[0] / (float)BQ;
}

// ---------------------------------------------------------------------------
extern "C" void kernel_launch(void* const* d_in, const int* in_sizes, int n_in,
                              void* d_out, int out_size, void* d_ws,
                              size_t ws_size, hipStream_t stream) {
  (void)in_sizes; (void)n_in; (void)out_size; (void)ws_size;
  const int* center_id = (const int*)d_in[0];
  const int* ctx_ids = (const int*)d_in[1];
  const float* eps = (const float*)d_in[2];
  const float* emb = (const float*)d_in[3];
  const float* prior_means = (const float*)d_in[4];
  const float* prior_vars = (const float*)d_in[5];
  const float* W_enc = (const float*)d_in[6];
  const float* b_enc = (const float*)d_in[7];
  const float* W_mean = (const float*)d_in[8];
  const float* b_mean = (const float*)d_in[9];
  const float* W_var = (const float*)d_in[10];
  const float* b_var = (const float*)d_in[11];
  const float* W_vocab = (const float*)d_in[12];
  const float* b_vocab = (const float*)d_in[13];
  float* out = (float*)d_out;

  // Workspace layout (floats): h | z | kl | part_m | part_s | resid  (~1.8 MB)
  float* ws = (float*)d_ws;
  float* h = ws;                              // B*256
  float* zbuf = h + (size_t)BQ * TWO_D;       // B*128
  float* kl = zbuf + (size_t)BQ * DIM;        // B
  float* part_m = kl + BQ;                    // NSLICE*B
  float* part_s = part_m + (size_t)NSLICE * BQ;
  float* resid = part_s + (size_t)NSLICE * BQ;

  enc_kernel<<<BQ, 512, 0, stream>>>(center_id, ctx_ids, emb, W_enc, b_enc, h);
  latent_kernel<<<BQ, 128, 0, stream>>>(center_id, eps, prior_means, prior_vars,
                                        W_mean, b_mean, W_var, b_var, h, zbuf, kl);
  vocab_lse_kernel<<<dim3(BQ / 16, NSLICE), 256, 0, stream>>>(zbuf, W_vocab,
                                                              b_vocab, part_m,
                                                              part_s);
  rec_kernel<<<BQ, 32, 0, stream>>>(ctx_ids, zbuf, W_vocab, b_vocab, part_m,
                                    part_s, kl, resid);
  mean_kernel<<<1, 256, 0, stream>>>(resid, out);
}